// GraphCNN_12962211299360
// MI455X (gfx1250) — compile-verified
//
#include <hip/hip_runtime.h>

#define N_NODES 50000
#define N_EDGES 800000
#define N_GRAPHS 128
#define DIN 200
#define DH 128
#define NL 4
#define OUTD (DIN + NL * DH)   // 712
#define BN_EPS 1e-5f

typedef __attribute__((ext_vector_type(2))) float v2f;
typedef __attribute__((ext_vector_type(8))) float v8f;

// ---------------- pooled = (1 + eps[l]) * h ----------------
__global__ void init_pooled_kernel(const float* __restrict__ h,
                                   float* __restrict__ pooled,
                                   const float* __restrict__ eps, int layer,
                                   int total) {
    float s = 1.0f + eps[layer];
    for (int i = blockIdx.x * blockDim.x + threadIdx.x; i < total;
         i += gridDim.x * blockDim.x)
        pooled[i] = s * h[i];
}

// ---------------- pooled[row] += h[col] over all edges ----------------
// One wave per edge; lanes stride the feature dimension (coalesced).
__global__ void spmm_scatter_kernel(const int* __restrict__ erow,
                                    const int* __restrict__ ecol,
                                    const float* __restrict__ h,
                                    float* __restrict__ pooled, int D) {
    int lane = threadIdx.x & 31;
    int wave = (blockIdx.x * blockDim.x + threadIdx.x) >> 5;
    int nwaves = (gridDim.x * blockDim.x) >> 5;
    for (int e = wave; e < N_EDGES; e += nwaves) {
        int r = erow[e];
        int c = ecol[e];
        const float* __restrict__ src = h + (size_t)c * D;
        float* __restrict__ dst = pooled + (size_t)r * D;
        for (int f = lane; f < D; f += 32)
            atomicAdd(&dst[f], src[f]);
    }
}

// ---------------- C[N x DH] = A[N x K] @ W[K x DH] + bias ----------------
// One wave computes a 16x64 tile with 4 f32 WMMA accumulators.
// grid = (N/16, DH/64), block = 32 threads (EXEC all ones for WMMA).
__global__ void gemm_bias_wmma_kernel(const float* __restrict__ A,
                                      const float* __restrict__ W,
                                      const float* __restrict__ bias,
                                      float* __restrict__ C, int K) {
    int lane = threadIdx.x;
    int m0 = blockIdx.x * 16;
    int c0 = blockIdx.y * 64;
    int halfsel = (lane >= 16) ? 1 : 0;   // lane half selects K offset 0 / 2
    int lr = lane & 15;

    v8f acc0 = {}, acc1 = {}, acc2 = {}, acc3 = {};

    // A fragment base: row = m0 + lr, K offset halfsel*2 (16x4 f32 layout)
    const float* __restrict__ arow = A + (size_t)(m0 + lr) * K + halfsel * 2;

    for (int k = 0; k < K; k += 4) {
        // b64 load: A[row][k + halfsel*2 .. +1]  (8B aligned: K,k,halfsel*2 even)
        v2f a = *(const v2f*)(arow + k);

        int kb = k + halfsel * 2;
        const float* __restrict__ wp = W + (size_t)kb * DH + c0 + lr;
        v2f b0, b1, b2, b3;
        b0.x = wp[0];   b0.y = wp[DH];
        b1.x = wp[16];  b1.y = wp[DH + 16];
        b2.x = wp[32];  b2.y = wp[DH + 32];
        b3.x = wp[48];  b3.y = wp[DH + 48];

        acc0 = __builtin_amdgcn_wmma_f32_16x16x4_f32(false, a, false, b0,
                                                     (short)0, acc0, false, false);
        acc1 = __builtin_amdgcn_wmma_f32_16x16x4_f32(false, a, false, b1,
                                                     (short)0, acc1, false, false);
        acc2 = __builtin_amdgcn_wmma_f32_16x16x4_f32(false, a, false, b2,
                                                     (short)0, acc2, false, false);
        acc3 = __builtin_amdgcn_wmma_f32_16x16x4_f32(false, a, false, b3,
                                                     (short)0, acc3, false, false);
    }

    // C/D layout: reg r -> row m0 + r (lanes 0-15) or m0 + 8 + r (lanes 16-31)
    int rbase = m0 + halfsel * 8;
    float bias0 = bias[c0 + lr];
    float bias1 = bias[c0 + lr + 16];
    float bias2 = bias[c0 + lr + 32];
    float bias3 = bias[c0 + lr + 48];
#pragma unroll
    for (int r = 0; r < 8; ++r) {
        float* __restrict__ op = C + (size_t)(rbase + r) * DH + c0 + lr;
        op[0]  = acc0[r] + bias0;
        op[16] = acc1[r] + bias1;
        op[32] = acc2[r] + bias2;
        op[48] = acc3[r] + bias3;
    }
}

// ---------------- per-column sum / sum-of-squares (double accumulation) ----
__global__ void bn_stats_kernel(const float* __restrict__ X,
                                double* __restrict__ stats) {
    __shared__ double s_sum[256];
    __shared__ double s_sq[256];
    int t = threadIdx.x;
    int col = t & (DH - 1);
    int rofs = t >> 7;  // 0 or 1
    int rowsPer = (N_NODES + gridDim.x - 1) / gridDim.x;
    int r0 = blockIdx.x * rowsPer;
    int r1 = r0 + rowsPer;
    if (r1 > N_NODES) r1 = N_NODES;
    double s = 0.0, q = 0.0;
    for (int r = r0 + rofs; r < r1; r += 2) {
        float v = X[(size_t)r * DH + col];
        s += (double)v;
        q += (double)v * (double)v;
    }
    s_sum[t] = s;
    s_sq[t] = q;
    __syncthreads();
    if (t < DH) {
        s = s_sum[t] + s_sum[t + DH];
        q = s_sq[t] + s_sq[t + DH];
        atomicAdd(&stats[col], s);
        atomicAdd(&stats[DH + col], q);
    }
}

// ---------------- turn stats into per-column scale/shift ----------------
__global__ void bn_finalize_kernel(const double* __restrict__ stats,
                                   const float* __restrict__ gamma,
                                   const float* __restrict__ beta,
                                   float* __restrict__ scaleshift) {
    int c = threadIdx.x;
    if (c < DH) {
        double mean = stats[c] / (double)N_NODES;
        double var = stats[DH + c] / (double)N_NODES - mean * mean;
        float inv = rsqrtf((float)var + BN_EPS);
        float sc = gamma[c] * inv;
        scaleshift[c] = sc;
        scaleshift[DH + c] = beta[c] - (float)mean * sc;
    }
}

// ---------------- x = relu(x * scale[c] + shift[c]) (in place) ----------
__global__ void bn_relu_kernel(float* __restrict__ X,
                               const float* __restrict__ scaleshift, int total) {
    for (int i = blockIdx.x * blockDim.x + threadIdx.x; i < total;
         i += gridDim.x * blockDim.x) {
        int c = i & (DH - 1);
        float v = X[i] * scaleshift[c] + scaleshift[DH + c];
        X[i] = v > 0.0f ? v : 0.0f;
    }
}

// ---------------- out[gid[node], colbase + f] += h[node, f] ----------------
__global__ void graph_pool_kernel(const float* __restrict__ h,
                                  const int* __restrict__ gid,
                                  float* __restrict__ out, int D, int colbase,
                                  int total) {
    for (int i = blockIdx.x * blockDim.x + threadIdx.x; i < total;
         i += gridDim.x * blockDim.x) {
        int node = i / D;
        int f = i - node * D;
        atomicAdd(&out[(size_t)gid[node] * OUTD + colbase + f], h[i]);
    }
}

extern "C" void kernel_launch(void* const* d_in, const int* in_sizes, int n_in,
                              void* d_out, int out_size, void* d_ws,
                              size_t ws_size, hipStream_t stream) {
    (void)in_sizes; (void)n_in; (void)ws_size;

    const float* x        = (const float*)d_in[0];
    const int*   erow     = (const int*)d_in[1];
    const int*   ecol     = (const int*)d_in[2];
    const int*   gid      = (const int*)d_in[3];
    const float* eps      = (const float*)d_in[4];
    const float* w1_0     = (const float*)d_in[5];
    const float* b1_0     = (const float*)d_in[6];
    const float* g1_0     = (const float*)d_in[7];
    const float* be1_0    = (const float*)d_in[8];
    const float* w2_0     = (const float*)d_in[9];
    const float* b2_0     = (const float*)d_in[10];
    const float* gbn_0    = (const float*)d_in[11];
    const float* bbn_0    = (const float*)d_in[12];
    const float* w1_r     = (const float*)d_in[13];
    const float* b1_r     = (const float*)d_in[14];
    const float* g1_r     = (const float*)d_in[15];
    const float* be1_r    = (const float*)d_in[16];
    const float* w2_r     = (const float*)d_in[17];
    const float* b2_r     = (const float*)d_in[18];
    const float* gbn_r    = (const float*)d_in[19];
    const float* bbn_r    = (const float*)d_in[20];

    float* out = (float*)d_out;

    // ---- workspace carve-up ----
    char* ws = (char*)d_ws;
    float*  pooled = (float*)ws;                                    // N*DIN f32
    float*  mid    = (float*)(ws + (size_t)N_NODES * DIN * 4);      // N*DH f32
    float*  hbuf   = (float*)(ws + (size_t)N_NODES * (DIN + DH) * 4);        // N*DH
    double* stats  = (double*)(ws + (size_t)N_NODES * (DIN + 2 * DH) * 4);   // 256 f64
    float*  scsh   = (float*)((char*)stats + 256 * sizeof(double)); // 256 f32

    const int TPB = 256;
    const int GS_BLOCKS = 4096;  // grid-stride elementwise kernels
    const int SPMM_BLOCKS = 2048;

    // zero output accumulator
    hipMemsetAsync(d_out, 0, (size_t)out_size * sizeof(float), stream);

    // pool the raw input features into columns [0, DIN)
    graph_pool_kernel<<<GS_BLOCKS, TPB, 0, stream>>>(x, gid, out, DIN, 0,
                                                     N_NODES * DIN);

    for (int l = 0; l < NL; ++l) {
        const float* h_in = (l == 0) ? x : hbuf;
        int Dl = (l == 0) ? DIN : DH;
        const float* w1  = (l == 0) ? w1_0  : w1_r  + (size_t)(l - 1) * DH * DH;
        const float* b1  = (l == 0) ? b1_0  : b1_r  + (size_t)(l - 1) * DH;
        const float* g1  = (l == 0) ? g1_0  : g1_r  + (size_t)(l - 1) * DH;
        const float* be1 = (l == 0) ? be1_0 : be1_r + (size_t)(l - 1) * DH;
        const float* w2  = (l == 0) ? w2_0  : w2_r  + (size_t)(l - 1) * DH * DH;
        const float* b2  = (l == 0) ? b2_0  : b2_r  + (size_t)(l - 1) * DH;
        const float* gbn = (l == 0) ? gbn_0 : gbn_r + (size_t)(l - 1) * DH;
        const float* bbn = (l == 0) ? bbn_0 : bbn_r + (size_t)(l - 1) * DH;

        // pooled = (1+eps)*h ; then scatter-add neighbors
        init_pooled_kernel<<<GS_BLOCKS, TPB, 0, stream>>>(h_in, pooled, eps, l,
                                                          N_NODES * Dl);
        spmm_scatter_kernel<<<SPMM_BLOCKS, TPB, 0, stream>>>(erow, ecol, h_in,
                                                             pooled, Dl);

        // MLP hidden: mid = pooled @ w1 + b1 ; BN ; ReLU
        gemm_bias_wmma_kernel<<<dim3(N_NODES / 16, DH / 64), 32, 0, stream>>>(
            pooled, w1, b1, mid, Dl);
        hipMemsetAsync(stats, 0, 256 * sizeof(double), stream);
        bn_stats_kernel<<<128, TPB, 0, stream>>>(mid, stats);
        bn_finalize_kernel<<<1, DH, 0, stream>>>(stats, g1, be1, scsh);
        bn_relu_kernel<<<GS_BLOCKS, TPB, 0, stream>>>(mid, scsh, N_NODES * DH);

        // MLP output: hbuf = mid @ w2 + b2 ; BN ; ReLU
        gemm_bias_wmma_kernel<<<dim3(N_NODES / 16, DH / 64), 32, 0, stream>>>(
            mid, w2, b2, hbuf, DH);
        hipMemsetAsync(stats, 0, 256 * sizeof(double), stream);
        bn_stats_kernel<<<128, TPB, 0, stream>>>(hbuf, stats);
        bn_finalize_kernel<<<1, DH, 0, stream>>>(stats, gbn, bbn, scsh);
        bn_relu_kernel<<<GS_BLOCKS, TPB, 0, stream>>>(hbuf, scsh, N_NODES * DH);

        // pool this layer's representation into columns [DIN + l*DH, ...)
        graph_pool_kernel<<<GS_BLOCKS, TPB, 0, stream>>>(
            hbuf, gid, out, DH, DIN + l * DH, N_NODES * DH);
    }
}